// ModelNew_17514876633481
// MI455X (gfx1250) — compile-verified
//
#include <hip/hip_runtime.h>
#include <cmath>

typedef __attribute__((ext_vector_type(2))) float v2f;
typedef __attribute__((ext_vector_type(8))) float v8f;

#define TILE_M 128
#define TILE_N 128
#define TILE_K 16
#define A_STRIDE 18      // 16 K-floats + 2 pad -> conflict-free b64 fragment reads
#define B_PSTRIDE 288    // words per k-pair row: 128 cols x 2 + 32 pad
                         // (+288 == +32 mod 64 banks -> halves of wave disjoint)

__device__ __forceinline__ void load_tile_regs(
    const float* __restrict__ A, const float* __restrict__ B,
    int n, int row0, int col0, int k0, int tid,
    v2f a_regs[4], float4 b_regs[2]) {
    #pragma unroll
    for (int i = 0; i < 4; ++i) {            // A: 128x16 floats as 1024 float2
        int e  = tid + i * 256;
        int r  = e >> 3;                      // 8 float2 per row
        int kk = (e & 7) << 1;
        a_regs[i] = *(const v2f*)(A + (size_t)(row0 + r) * n + k0 + kk);
    }
    #pragma unroll
    for (int i = 0; i < 2; ++i) {            // B: 16x128 floats as 512 float4
        int e  = tid + i * 256;
        int r  = e >> 5;                      // 32 float4 per row
        int cc = (e & 31) << 2;
        b_regs[i] = *(const float4*)(B + (size_t)(k0 + r) * n + col0 + cc);
    }
}

__device__ __forceinline__ void store_tile_lds(
    float* __restrict__ Abuf, float* __restrict__ Bbuf, int tid,
    const v2f a_regs[4], const float4 b_regs[2]) {
    #pragma unroll
    for (int i = 0; i < 4; ++i) {
        int e  = tid + i * 256;
        int r  = e >> 3;
        int kk = (e & 7) << 1;
        *(v2f*)(Abuf + r * A_STRIDE + kk) = a_regs[i];
    }
    #pragma unroll
    for (int i = 0; i < 2; ++i) {
        int e  = tid + i * 256;
        int r  = e >> 5;
        int cc = (e & 31) << 2;
        // k-pair interleaved: Bp[r/2][col][r%2]
        float* base = Bbuf + (r >> 1) * B_PSTRIDE + (r & 1);
        const float4 v = b_regs[i];
        base[(cc + 0) << 1] = v.x;
        base[(cc + 1) << 1] = v.y;
        base[(cc + 2) << 1] = v.z;
        base[(cc + 3) << 1] = v.w;
    }
}

__global__ __launch_bounds__(256)
void trimm_wmma_f32(const float* __restrict__ A, const float* __restrict__ B,
                    float* __restrict__ C, int n) {
    __shared__ float Atile[2][TILE_M * A_STRIDE];   // 2 x 9.0 KB
    __shared__ float Btile[2][8 * B_PSTRIDE];       // 2 x 9.0 KB

    const int nb  = n / TILE_M;
    const int bm  = blockIdx.x / nb;
    const int bn  = blockIdx.x % nb;
    const int tid = threadIdx.x;

    const int row0 = bm * TILE_M;
    const int col0 = bn * TILE_N;

    if (bn > bm) {
        // Strictly upper-triangular tile: exact zeros (tril inputs guarantee it).
        const float4 z = make_float4(0.f, 0.f, 0.f, 0.f);
        #pragma unroll
        for (int i = 0; i < 16; ++i) {
            int e  = tid + i * 256;
            int r  = e >> 5;
            int cc = (e & 31) << 2;
            *(float4*)(C + (size_t)(row0 + r) * n + col0 + cc) = z;
        }
        return;
    }

    const int lane   = tid & 31;
    const int wave   = tid >> 5;
    const int wave_m = wave & 3;   // 0..3 -> 32 rows each
    const int wave_n = wave >> 2;  // 0..1 -> 64 cols each
    const int hi     = lane >> 4;  // half-wave: A/B hold K+2, C/D hold M+8
    const int l16    = lane & 15;

    v8f acc[2][4] = {};

    // Preload first K-tile (kb = bn) into buffer 0.
    {
        v2f a_regs[4]; float4 b_regs[2];
        load_tile_regs(A, B, n, row0, col0, bn * TILE_K, tid, a_regs, b_regs);
        store_tile_lds(Atile[0], Btile[0], tid, a_regs, b_regs);
    }
    __syncthreads();

    int cur = 0;
    for (int kb = bn; kb <= bm; ++kb) {
        const bool has_next = (kb < bm);

        // Prefetch next K-tile into registers (overlaps with WMMA below).
        v2f a_regs[4]; float4 b_regs[2];
        if (has_next)
            load_tile_regs(A, B, n, row0, col0, (kb + 1) * TILE_K, tid,
                           a_regs, b_regs);

        const float* __restrict__ Ab = Atile[cur];
        const float* __restrict__ Bb = Btile[cur];

        #pragma unroll
        for (int ks = 0; ks < 4; ++ks) {
            const int kk = ks * 4 + hi * 2;  // even; fragment K-pair base
            const int kp = kk >> 1;

            v2f afrag[2];
            #pragma unroll
            for (int sm = 0; sm < 2; ++sm) {
                int r = wave_m * 32 + sm * 16 + l16;
                afrag[sm] = *(const v2f*)(Ab + r * A_STRIDE + kk);
            }

            v2f bfrag[4];
            #pragma unroll
            for (int sn = 0; sn < 4; ++sn) {
                int c = wave_n * 64 + sn * 16 + l16;
                bfrag[sn] = *(const v2f*)(Bb + kp * B_PSTRIDE + (c << 1));
            }

            #pragma unroll
            for (int sm = 0; sm < 2; ++sm)
                #pragma unroll
                for (int sn = 0; sn < 4; ++sn)
                    acc[sm][sn] = __builtin_amdgcn_wmma_f32_16x16x4_f32(
                        /*neg_a=*/false, afrag[sm],
                        /*neg_b=*/false, bfrag[sn],
                        /*c_mod=*/(short)0, acc[sm][sn],
                        /*reuse_a=*/false, /*reuse_b=*/false);
        }

        // Drain prefetched tile into the other buffer, then one barrier.
        if (has_next)
            store_tile_lds(Atile[cur ^ 1], Btile[cur ^ 1], tid, a_regs, b_regs);
        __syncthreads();
        cur ^= 1;
    }

    // Write out: VGPR v of a 16x16 f32 D-tile holds row (v + 8*hi), col l16.
    #pragma unroll
    for (int sm = 0; sm < 2; ++sm)
        #pragma unroll
        for (int sn = 0; sn < 4; ++sn)
            #pragma unroll
            for (int v = 0; v < 8; ++v) {
                int r = row0 + wave_m * 32 + sm * 16 + hi * 8 + v;
                int c = col0 + wave_n * 64 + sn * 16 + l16;
                C[(size_t)r * n + c] = acc[sm][sn][v];
            }
}

extern "C" void kernel_launch(void* const* d_in, const int* in_sizes, int n_in,
                              void* d_out, int out_size, void* d_ws, size_t ws_size,
                              hipStream_t stream) {
    const float* A = (const float*)d_in[0];
    const float* B = (const float*)d_in[1];
    float* C = (float*)d_out;

    int n = (int)lroundf(sqrtf((float)in_sizes[0]));  // 4096
    int nb = n / TILE_M;                              // 32
    dim3 grid(nb * nb);
    dim3 block(256);
    trimm_wmma_f32<<<grid, block, 0, stream>>>(A, B, C, n);
}